// DeepSeekV3MoeRoutingMethod_66340064854661
// MI455X (gfx1250) — compile-verified
//
#include <hip/hip_runtime.h>
#include <math.h>

// DeepSeek-V3 MoE routing for MI455X (gfx1250, wave32).
// One token (256 experts) per wave; lane l owns experts [8l, 8l+8).
// Logit rows are DMA'd into LDS with the CDNA5 Tensor Data Mover.

typedef unsigned int uint32x4 __attribute__((ext_vector_type(4)));
typedef int          int32x8  __attribute__((ext_vector_type(8)));
typedef int          int32x4  __attribute__((ext_vector_type(4)));

#define WAVE 32

__device__ __forceinline__ float sigmoid_f(float x) {
    return 1.0f / (1.0f + expf(-x));
}

__global__ __launch_bounds__(256) void moe_route_kernel(
    const float* __restrict__ logits,   // [T, 256]
    const float* __restrict__ bias,     // [256]
    int*   __restrict__ out_idx,        // [T, 8]
    float* __restrict__ out_val,        // [T, 8]
    int T)
{
    __shared__ __align__(16) float lrow[8 * 256];   // 1 KB per wave
    const int lane  = threadIdx.x & 31;
    const int wave  = threadIdx.x >> 5;
    const int token = blockIdx.x * 8 + wave;
    if (token >= T) return;

    float* myrow = &lrow[wave * 256];

    // ---------------- Stage the 1KB logit row into LDS ----------------
#if __has_builtin(__builtin_amdgcn_tensor_load_to_lds)
    {
        // D# group0: count=1 (user), lds_addr, 57-bit global addr, type=2
        unsigned lds_off = (unsigned)(unsigned long long)myrow;   // low 32 bits = LDS byte offset
        unsigned long long ga =
            (unsigned long long)(const void*)(logits + (size_t)token * 256);
        uint32x4 g0;
        g0[0] = 1u;                                          // count=1, is_restore=0, gather off
        g0[1] = lds_off;                                     // LDS byte address
        g0[2] = (unsigned)(ga & 0xffffffffull);              // global_addr[31:0]
        g0[3] = (unsigned)((ga >> 32) & 0x01fffffful)        // global_addr[56:32]
              | 0x80000000u;                                 // type = 2 ("image")
        // D# group1: data_size=4B, tensor_dim0=256, tensor_dim1=1,
        //            tile_dim0=256, tile_dim1=1, dim0 stride=256, mask=0 (no cluster)
        int32x8 g1;
        g1[0] = 2 << 16;        // workgroup_mask=0, data_size=2 (4 bytes)
        g1[1] = 256 << 16;      // atomic_barrier_addr=0, tensor_dim0[15:0]=256
        g1[2] = 1 << 16;        // tensor_dim0[31:16]=0, tensor_dim1[15:0]=1
        g1[3] = 256 << 16;      // tensor_dim1[31:16]=0, tile_dim0=256
        g1[4] = 1;              // tile_dim1=1, tile_dim2=0
        g1[5] = 256;            // tensor_dim0_stride[31:0]=256 elements
        g1[6] = 0;              // stride hi, tensor_dim1_stride lo
        g1[7] = 0;              // tensor_dim1_stride hi
        int32x4 g2 = {0, 0, 0, 0};
        int32x4 g3 = {0, 0, 0, 0};
        int32x8 g4 = {0, 0, 0, 0, 0, 0, 0, 0};   // unused VADDR4 slot (clang-23 6-arg form)
        __builtin_amdgcn_tensor_load_to_lds(g0, g1, g2, g3, g4, 0);
        __builtin_amdgcn_s_wait_tensorcnt(0);
        asm volatile("" ::: "memory");   // LDS now holds the row; don't hoist ds loads
    }
#else
    {
        const float4* g4p =
            reinterpret_cast<const float4*>(logits + (size_t)token * 256 + lane * 8);
        float4 a = g4p[0], b = g4p[1];
        float4* s4 = reinterpret_cast<float4*>(myrow + lane * 8);
        s4[0] = a; s4[1] = b;
        asm volatile("" ::: "memory");
    }
#endif

    // ---------------- Per-lane: 8 logits from LDS + 8 bias ----------------
    float x[8], bz[8];
    {
        const float4* l4 = reinterpret_cast<const float4*>(myrow + lane * 8);
        float4 a = l4[0], b = l4[1];
        x[0]=a.x; x[1]=a.y; x[2]=a.z; x[3]=a.w;
        x[4]=b.x; x[5]=b.y; x[6]=b.z; x[7]=b.w;
        const float4* b4 = reinterpret_cast<const float4*>(bias + lane * 8);
        float4 c = b4[0], d = b4[1];
        bz[0]=c.x; bz[1]=c.y; bz[2]=c.z; bz[3]=c.w;
        bz[4]=d.x; bz[5]=d.y; bz[6]=d.z; bz[7]=d.w;
    }

    float swb[8];
#pragma unroll
    for (int i = 0; i < 8; ++i) swb[i] = sigmoid_f(x[i]) + bz[i];

    // ---------------- Group score: top-2 sum within quad (32 experts) -----
    float m1 = fmaxf(swb[0], swb[1]);
    float m2 = fminf(swb[0], swb[1]);
#pragma unroll
    for (int i = 2; i < 8; ++i) {
        m2 = fmaxf(m2, fminf(swb[i], m1));
        m1 = fmaxf(m1, swb[i]);
    }
#pragma unroll
    for (int off = 1; off <= 2; off <<= 1) {
        float p1 = __shfl_xor(m1, off, WAVE);
        float p2 = __shfl_xor(m2, off, WAVE);
        m2 = fmaxf(fminf(m1, p1), fmaxf(m2, p2));
        m1 = fmaxf(m1, p1);
    }
    float gscore = m1 + m2;            // identical across the 4 lanes of a group

    // ---------------- Select top-4 groups (tie -> lower group index) ------
    const int myg = lane >> 2;
    int better = 0;
#pragma unroll
    for (int g = 0; g < 8; ++g) {
        float gs = __shfl(gscore, g * 4, WAVE);
        better += (gs > gscore) ? 1 : 0;
        better += ((gs == gscore) && (g < myg)) ? 1 : 0;
    }
    const bool sel = (better < 4);

    // ---------------- 8 rounds of wave argmax over masked biased scores ---
    const float NEG = -__builtin_inff();
    float v[8];
#pragma unroll
    for (int i = 0; i < 8; ++i) v[i] = sel ? swb[i] : NEG;

    int pick = 0;   // after the loop: lane r (r<8) holds expert id of r-th pick
#pragma unroll
    for (int r = 0; r < 8; ++r) {
        float mloc = v[0];
#pragma unroll
        for (int i = 1; i < 8; ++i) mloc = fmaxf(mloc, v[i]);
        float w = mloc;
#pragma unroll
        for (int off = 16; off >= 1; off >>= 1)
            w = fmaxf(w, __shfl_xor(w, off, WAVE));

        int slot = 0;
#pragma unroll
        for (int i = 7; i >= 0; --i) {           // lowest slot wins on (rare) ties
            bool c = (v[i] == w);
            v[i]  = c ? NEG : v[i];              // remove winner
            slot  = c ? i : slot;
        }
        unsigned long long bal = __ballot(mloc == w);
        int L   = (int)__builtin_ctzll(bal);     // lowest lane -> lowest expert idx
        int eid = __builtin_amdgcn_readlane((lane << 3) | slot, L);  // SGPR
        pick = (lane == r) ? eid : pick;
    }

    // ------- Gather winners' unbiased scores from LDS, renormalize --------
    float score = sigmoid_f(myrow[pick]);        // 1 ds_load_b32; lanes >=8 unused
    float ssum = score;
#pragma unroll
    for (int off = 1; off <= 4; off <<= 1)       // sum over lanes 0..7
        ssum += __shfl_xor(ssum, off, WAVE);
    float val = score / (ssum + 1e-20f) * 2.5f;

    // ------- Bitonic sort 8 (lanes 0..7): descending val, tie -> low idx --
    int   vi = pick;
    float vv = val;
    const int KS[6] = {2, 4, 4, 8, 8, 8};
    const int JS[6] = {1, 2, 1, 4, 2, 1};
#pragma unroll
    for (int s = 0; s < 6; ++s) {
        int k = KS[s], j = JS[s];
        float ov = __shfl_xor(vv, j, WAVE);
        int   oi = __shfl_xor(vi, j, WAVE);
        bool dirDesc     = (lane & k) == 0;
        bool lowerLane   = (lane & j) == 0;
        bool wantMax     = (lowerLane == dirDesc);
        bool otherBetter = (ov > vv) || ((ov == vv) && (oi < vi));
        bool take        = (otherBetter == wantMax);
        vv = take ? ov : vv;
        vi = take ? oi : vi;
    }

    if (lane < 8) {
        out_idx[(size_t)token * 8 + lane] = vi;
        out_val[(size_t)token * 8 + lane] = vv;
    }
}

extern "C" void kernel_launch(void* const* d_in, const int* in_sizes, int n_in,
                              void* d_out, int out_size, void* d_ws, size_t ws_size,
                              hipStream_t stream) {
    const float* logits = (const float*)d_in[0];
    const float* bias   = (const float*)d_in[1];
    const int E = in_sizes[1];                   // 256
    const int T = in_sizes[0] / E;               // 131072
    int*   oidx = (int*)d_out;                   // [T,8] int32
    float* oval = (float*)d_out + (size_t)T * 8; // [T,8] float32
    const int blocks = (T + 7) / 8;              // 8 tokens (waves) per block
    moe_route_kernel<<<blocks, 256, 0, stream>>>(logits, bias, oidx, oval, T);
}